// KnowledgeD2V_6622839571289
// MI455X (gfx1250) — compile-verified
//
#include <hip/hip_runtime.h>
#include <hip/hip_bf16.h>

#define BATCH   16384
#define WINDOW  21
#define EMBED   128
#define NSAMP   10
#define KTOT    (NSAMP + 1)
#define ROWS_PER_WAVE   16
#define WAVES_PER_BLOCK 4
#define GROUPS (BATCH / ROWS_PER_WAVE)            // 1024
#define BLOCKS (GROUPS / WAVES_PER_BLOCK)         // 256

typedef __attribute__((ext_vector_type(2))) float v2f;
typedef __attribute__((ext_vector_type(8))) float v8f;

static_assert(BATCH % (ROWS_PER_WAVE * WAVES_PER_BLOCK) == 0, "even tiling");

// log_sigmoid(z) = min(z,0) - log1p(exp(-|z|))  (numerically stable)
__device__ __forceinline__ float log_sigmoid(float z) {
    return fminf(z, 0.0f) - log1pf(expf(-fabsf(z)));
}

__global__ __launch_bounds__(WAVES_PER_BLOCK * 32)
void d2v_main(const int* __restrict__ input_labels,   // [BATCH, WINDOW]
              const int* __restrict__ out_labels,     // [BATCH]
              const int* __restrict__ noise,          // [BATCH, NSAMP]
              const float* __restrict__ word_embed,   // [NUM_WORDS, EMBED]
              const float* __restrict__ out_embed,    // [NUM_WORDS, EMBED]
              const float* __restrict__ doc_embed,    // [NUM_DOCS, EMBED]
              float* __restrict__ partial)            // [GROUPS]
{
    __shared__ float xtile[WAVES_PER_BLOCK][ROWS_PER_WAVE * EMBED]; // 4 x 8KB
    __shared__ float stile[WAVES_PER_BLOCK][16 * 16];               // 4 x 1KB

    const int lane  = threadIdx.x & 31;
    const int wave  = threadIdx.x >> 5;
    const int group = blockIdx.x * WAVES_PER_BLOCK + wave;  // 0..GROUPS-1
    const int row0  = group * ROWS_PER_WAVE;

    float* X = xtile[wave];
    float* S = stile[wave];

    // ---- Phase 1: build X rows (coalesced 512B gathers, float4 per lane) ----
    for (int r = 0; r < ROWS_PER_WAVE; ++r) {
        const int b = row0 + r;
        const int* lab = input_labels + (size_t)b * WINDOW;
        const int doc = lab[WINDOW - 1];
        float4 acc = ((const float4*)(doc_embed + (size_t)doc * EMBED))[lane];
        for (int w = 0; w < WINDOW - 1; ++w) {
            const int cw = lab[w];
            float4 v = ((const float4*)(word_embed + (size_t)cw * EMBED))[lane];
            acc.x += v.x; acc.y += v.y; acc.z += v.z; acc.w += v.w;
        }
        ((float4*)(X + r * EMBED))[lane] = acc;   // ds_store_b128
    }
    // LDS ops within a wave are kept in order by hardware; subsequent
    // ds_loads from X by other lanes of this wave see these stores.

    // WMMA A/B lane addressing:
    //   A 16x4 f32:  VGPR0 = {K=0 lanes0-15, K=2 lanes16-31}, VGPR1 = {K=1, K=3}
    //   B 4x16 f32:  same K striping, N = lane&15
    const int rsel  = lane & 15;          // A row / B column (= batch row in group)
    const int dhalf = (lane >> 4) * 2;    // 0 or 2

    float lrow = 0.0f;                    // per-row loss (lanes 0..15 valid)

    for (int k = 0; k < KTOT; ++k) {
        const int b_r = row0 + rsel;
        const int tid = (k == 0) ? out_labels[b_r]
                                 : noise[(size_t)b_r * NSAMP + (k - 1)];
        const float* yrow = out_embed + (size_t)tid * EMBED;

        v8f acc = {};
        #pragma unroll 4
        for (int c = 0; c < EMBED / 4; ++c) {
            const int d = 4 * c + dhalf;
            v2f a = *(const v2f*)(X + rsel * EMBED + d);   // ds_load_b64
            v2f bb = *(const v2f*)(yrow + d);              // global_load_b64
            // D[m,n] += sum_K A[m,K]*B[K,n] ; diagonal == scores for this k
            acc = __builtin_amdgcn_wmma_f32_16x16x4_f32(
                      false, a, false, bb, (short)0, acc, false, false);
        }

        // Spill 16x16 accumulator to LDS, read the diagonal.
        // acc[v]: lanes0-15 -> (M=v, N=lane); lanes16-31 -> (M=v+8, N=lane-16)
        const int mhi = (lane >> 4) << 3;
        #pragma unroll
        for (int v = 0; v < 8; ++v) {
            S[(v + mhi) * 16 + rsel] = acc[v];
        }
        const float s = S[rsel * 16 + rsel];   // score[row=rsel, k]
        const float z = (k == 0) ? s : -s;
        const float ls = log_sigmoid(z);
        if (lane < 16) lrow += ls;
    }

    // ---- Wave-level reduction over the 16 rows, fixed slot (deterministic) ----
    float val = (lane < 16) ? lrow : 0.0f;
    #pragma unroll
    for (int off = 16; off > 0; off >>= 1)
        val += __shfl_xor(val, off, 32);
    if (lane == 0) partial[group] = val;
}

__global__ __launch_bounds__(256)
void d2v_finalize(const float* __restrict__ partial, float* __restrict__ out)
{
    __shared__ float red[256];
    float s = 0.0f;
    for (int i = threadIdx.x; i < GROUPS; i += 256) s += partial[i];
    red[threadIdx.x] = s;
    __syncthreads();
    #pragma unroll
    for (int off = 128; off > 0; off >>= 1) {
        if (threadIdx.x < off) red[threadIdx.x] += red[threadIdx.x + off];
        __syncthreads();
    }
    if (threadIdx.x == 0) out[0] = -red[0] / (float)BATCH;
}

extern "C" void kernel_launch(void* const* d_in, const int* in_sizes, int n_in,
                              void* d_out, int out_size, void* d_ws, size_t ws_size,
                              hipStream_t stream)
{
    const int*   input_labels = (const int*)  d_in[0];   // [BATCH, WINDOW]
    const int*   out_labels   = (const int*)  d_in[1];   // [BATCH]
    const int*   noise        = (const int*)  d_in[2];   // [BATCH, NSAMP]
    // d_in[3] = num_sampled scalar (compile-time NSAMP)
    const float* word_embed   = (const float*)d_in[4];
    const float* out_embed    = (const float*)d_in[5];
    const float* doc_embed    = (const float*)d_in[6];
    float* out     = (float*)d_out;
    float* partial = (float*)d_ws;                       // GROUPS floats, fully rewritten every call

    d2v_main<<<BLOCKS, WAVES_PER_BLOCK * 32, 0, stream>>>(
        input_labels, out_labels, noise, word_embed, out_embed, doc_embed, partial);
    d2v_finalize<<<1, 256, 0, stream>>>(partial, out);
}